// MARN_58411555225770
// MI455X (gfx1250) — compile-verified
//
#include <hip/hip_runtime.h>
#include <math.h>

// ---------------- problem constants (from reference) ----------------
#define TSTEPS 128
#define NBATCH 1024
#define ROWS   16        // batch rows per workgroup (WMMA M)
#define DT 300
#define DA 74
#define DV 35
#define HT 128
#define HA 32
#define HV 32
#define TOT 192
#define KATT 4
// LDS row strides (odd to dodge 64-bank conflicts on A-fragment reads)
#define SXT 301
#define SXA 77          // pad: cols 74,75 zero (PK=76)
#define SXV 37          // pad: col 35 zero   (PK=36)
#define SHC 193
#define SZ  193
#define SAT 769
#define SSL 97

// ---------------- packed-B workspace layout (float offsets) ----------------
// Each weight [K,N] is packed into per-lane WMMA B-fragment order:
//   dst[((tile*(PK/4)+kg)*32 + lane)*2 + j] = src[(4*kg + 2*(lane>>4) + j)*N + 16*tile + (lane&15)]
// (zero-filled for k >= K), so one global_load_b64 per WMMA B fragment,
// lane-consecutive -> one coalesced 256B transaction.
#define OFF_WT  0u          // 512 x 300
#define OFF_UT  153600u     // 512 x 128
#define OFF_VT  219136u     // 512 x 192
#define OFF_WA  317440u     // 128 x 76
#define OFF_UA  327168u     // 128 x 32
#define OFF_VA  331264u     // 128 x 192
#define OFF_WV  355840u     // 128 x 36
#define OFF_UV  360448u     // 128 x 32
#define OFF_VV  364544u     // 128 x 192
#define OFF_AW  389120u     // 768 x 192
#define OFF_RT  536576u     // 64 x 512
#define OFF_RA  569344u     // 16 x 128
#define OFF_RV  571392u     // 16 x 128
#define OFF_HY  573440u     // 192 x 96
#define WS_FLOATS 591872u   // ~2.31 MB

typedef float v2f __attribute__((ext_vector_type(2)));
typedef float v8f __attribute__((ext_vector_type(8)));

__device__ __forceinline__ v8f wmma4(v2f a, v2f b, v8f c) {
  // D = A(16x4 f32) * B(4x16 f32) + C(16x16 f32)   -- CDNA5 V_WMMA_F32_16X16X4_F32
  return __builtin_amdgcn_wmma_f32_16x16x4_f32(false, a, false, b, (short)0, c, false, false);
}

__device__ __forceinline__ v8f bcast8(float x) {
  v8f r;
#pragma unroll
  for (int i = 0; i < 8; ++i) r[i] = x;
  return r;
}

__device__ __forceinline__ float sigm(float x) { return 1.0f / (1.0f + __expf(-x)); }

// A-column index mappers (realize kh.reshape views without materializing them)
struct IdMap { __device__ __forceinline__ int operator()(int k) const { return k; } };
struct MapT  { __device__ __forceinline__ int operator()(int q) const { return ((q >> 7) * TOT) + (q & 127); } };
struct MapA  { __device__ __forceinline__ int operator()(int q) const { return ((q >> 5) * TOT) + HT + (q & 31); } };
struct MapV  { __device__ __forceinline__ int operator()(int q) const { return ((q >> 5) * TOT) + HT + HA + (q & 31); } };

// All GEMM inner loops are software-pipelined (depth 2): the next k-group's
// A fragment (LDS) and B fragments (global, packed) are fetched into fresh
// registers BEFORE the current k-group's WMMAs issue, so the scheduler can
// overlap memory latency with matrix math instead of s_wait_loadcnt 0.

// ---- one 16x16 output tile: A from LDS (via mapper), B pre-packed ----
template <typename Map>
__device__ __forceinline__ void gemm1p(const float* __restrict__ As, int sA, Map map,
                                       const v2f* __restrict__ Bp, int nkg, int t0,
                                       int lane, int lrow, int half, v8f& A0) {
  const float* arow = As + lrow * sA;
  const v2f* b0p = Bp + (size_t)t0 * nkg * 32 + lane;
  v2f a;
  a.x = arow[map(half << 1)];
  a.y = arow[map((half << 1) + 1)];
  v2f b0 = b0p[0];
#pragma unroll 2
  for (int kg = 0; kg < nkg - 1; ++kg) {
    const int ka = ((kg + 1) << 2) + (half << 1);
    v2f an, n0;
    an.x = arow[map(ka)];
    an.y = arow[map(ka + 1)];
    n0 = b0p[(size_t)(kg + 1) * 32];
    A0 = wmma4(a, b0, A0);
    a = an; b0 = n0;
  }
  A0 = wmma4(a, b0, A0);
}

// ---- two tiles sharing one A fragment per k-step ----
template <typename Map>
__device__ __forceinline__ void gemm2p(const float* __restrict__ As, int sA, Map map,
                                       const v2f* __restrict__ Bp, int nkg, int t0, int t1,
                                       int lane, int lrow, int half, v8f& A0, v8f& A1) {
  const float* arow = As + lrow * sA;
  const v2f* b0p = Bp + (size_t)t0 * nkg * 32 + lane;
  const v2f* b1p = Bp + (size_t)t1 * nkg * 32 + lane;
  v2f a;
  a.x = arow[map(half << 1)];
  a.y = arow[map((half << 1) + 1)];
  v2f b0 = b0p[0], b1 = b1p[0];
#pragma unroll 2
  for (int kg = 0; kg < nkg - 1; ++kg) {
    const int ka = ((kg + 1) << 2) + (half << 1);
    v2f an, n0, n1;
    an.x = arow[map(ka)];
    an.y = arow[map(ka + 1)];
    n0 = b0p[(size_t)(kg + 1) * 32];
    n1 = b1p[(size_t)(kg + 1) * 32];
    A0 = wmma4(a, b0, A0);
    A1 = wmma4(a, b1, A1);
    a = an; b0 = n0; b1 = n1;
  }
  A0 = wmma4(a, b0, A0);
  A1 = wmma4(a, b1, A1);
}

// ---- four gate tiles (f,i,o,g) sharing one A fragment per k-step ----
template <typename Map>
__device__ __forceinline__ void gemm4p(const float* __restrict__ As, int sA, Map map,
                                       const v2f* __restrict__ Bp, int nkg,
                                       int t0, int t1, int t2, int t3,
                                       int lane, int lrow, int half,
                                       v8f& A0, v8f& A1, v8f& A2, v8f& A3) {
  const float* arow = As + lrow * sA;
  const v2f* b0p = Bp + (size_t)t0 * nkg * 32 + lane;
  const v2f* b1p = Bp + (size_t)t1 * nkg * 32 + lane;
  const v2f* b2p = Bp + (size_t)t2 * nkg * 32 + lane;
  const v2f* b3p = Bp + (size_t)t3 * nkg * 32 + lane;
  v2f a;
  a.x = arow[map(half << 1)];
  a.y = arow[map((half << 1) + 1)];
  v2f b0 = b0p[0], b1 = b1p[0], b2 = b2p[0], b3 = b3p[0];
#pragma unroll 2
  for (int kg = 0; kg < nkg - 1; ++kg) {
    const int ka = ((kg + 1) << 2) + (half << 1);
    v2f an, n0, n1, n2, n3;
    an.x = arow[map(ka)];
    an.y = arow[map(ka + 1)];
    n0 = b0p[(size_t)(kg + 1) * 32];
    n1 = b1p[(size_t)(kg + 1) * 32];
    n2 = b2p[(size_t)(kg + 1) * 32];
    n3 = b3p[(size_t)(kg + 1) * 32];
    A0 = wmma4(a, b0, A0);
    A1 = wmma4(a, b1, A1);
    A2 = wmma4(a, b2, A2);
    A3 = wmma4(a, b3, A3);
    a = an; b0 = n0; b1 = n1; b2 = n2; b3 = n3;
  }
  A0 = wmma4(a, b0, A0);
  A1 = wmma4(a, b1, A1);
  A2 = wmma4(a, b2, A2);
  A3 = wmma4(a, b3, A3);
}

// LSTHM cell: this wave owns 16 hidden columns (all four gates for those
// columns), so gate nonlinearities and cell state stay in registers.
__device__ __forceinline__ void lsthm_cell(
    const float* __restrict__ xs, int sX, int nkgX, const v2f* __restrict__ Wp,
    const float* __restrict__ hs, int nkgU, const v2f* __restrict__ Up,
    const float* __restrict__ zs, const v2f* __restrict__ Vp,
    const float* __restrict__ Wb, const float* __restrict__ Ub, const float* __restrict__ Vb,
    int H, int gcol, int lane, int lrow, int half,
    v8f& cstate, v8f& hout) {
  const int c0 = gcol, c1 = H + gcol, c2 = 2 * H + gcol, c3 = 3 * H + gcol;
  const int ht = H >> 4;                  // tiles per gate block
  const int t0 = gcol >> 4, t1 = t0 + ht, t2 = t0 + 2 * ht, t3 = t0 + 3 * ht;
  v8f F = bcast8(Wb[c0 + lrow] + Ub[c0 + lrow] + Vb[c0 + lrow]);
  v8f I = bcast8(Wb[c1 + lrow] + Ub[c1 + lrow] + Vb[c1 + lrow]);
  v8f O = bcast8(Wb[c2 + lrow] + Ub[c2 + lrow] + Vb[c2 + lrow]);
  v8f G = bcast8(Wb[c3 + lrow] + Ub[c3 + lrow] + Vb[c3 + lrow]);
  gemm4p(xs, sX,  IdMap(), Wp, nkgX,    t0, t1, t2, t3, lane, lrow, half, F, I, O, G);
  gemm4p(hs, SHC, IdMap(), Up, nkgU,    t0, t1, t2, t3, lane, lrow, half, F, I, O, G);
  gemm4p(zs, SZ,  IdMap(), Vp, TOT / 4, t0, t1, t2, t3, lane, lrow, half, F, I, O, G);
#pragma unroll
  for (int v = 0; v < 8; ++v) {
    float f = sigm(F[v]);
    float i = sigm(I[v]);
    float o = sigm(O[v]);
    float g = tanhf(G[v]);
    float cn = f * cstate[v] + i * g;
    cstate[v] = cn;
    hout[v] = tanhf(cn) * o;
  }
}

// ---------------- weight pre-pack kernel (runs every call; deterministic) ----------------
__global__ void pack_b_kernel(const float* __restrict__ src, float* __restrict__ dst,
                              int K, int N, int PK) {
  const int i = blockIdx.x * 256 + threadIdx.x;   // packed element index, [0, N*PK)
  if (i >= N * PK) return;
  const int j    = i & 1;
  const int lane = (i >> 1) & 31;
  const int nkg  = PK >> 2;
  const int kg   = (i >> 6) % nkg;
  const int tile = (i >> 6) / nkg;
  const int k    = (kg << 2) + ((lane >> 4) << 1) + j;
  const int col  = (tile << 4) + (lane & 15);
  dst[i] = (k < K) ? src[(size_t)k * N + col] : 0.0f;
}

struct P {
  const float *xt, *xa, *xv;
  const int* len;
  // packed weights (in d_ws)
  const v2f *Wt, *Ut, *Vt, *Wa, *Ua, *Va, *Wv, *Uv, *Vv, *AW, *Rt, *Ra, *Rv, *HY;
  // biases / epilogue (read directly from global)
  const float *Wb_t, *Ub_t, *Vb_t, *Rb_t;
  const float *Wb_a, *Ub_a, *Vb_a, *Rb_a;
  const float *Wb_v, *Ub_v, *Vb_v, *Rb_v;
  const float *att_b, *hyb_b, *out_W, *out_b;
  float* out;
};

__global__ __launch_bounds__(256, 1)
void marn_fused_kernel(P p) {
  __shared__ float xt[ROWS * SXT];
  __shared__ float xa[ROWS * SXA];
  __shared__ float xv[ROWS * SXV];
  __shared__ float hc[ROWS * SHC];      // concat(h_t,h_a,h_v) per row
  __shared__ float zb[ROWS * SZ];       // recurrent z
  __shared__ float att[ROWS * SAT];     // attention scores, then kh in place
  __shared__ float sloc[ROWS * SSL];    // concat(s_t,s_a,s_v) [96]
  __shared__ float zsel[ROWS * TOT];    // z at t = len-1
  __shared__ int   lens[ROWS];

  const int tid  = threadIdx.x;
  const int w    = tid >> 5;            // wave id (8 waves)
  const int lane = tid & 31;
  const int lrow = lane & 15;
  const int half = lane >> 4;
  const int rb   = blockIdx.x * ROWS;   // global batch-row base

  for (int i = tid; i < ROWS * SHC; i += 256) { hc[i] = 0.0f; zb[i] = 0.0f; }
  for (int i = tid; i < ROWS * SXA; i += 256) xa[i] = 0.0f;   // establish K pads
  for (int i = tid; i < ROWS * SXV; i += 256) xv[i] = 0.0f;
  for (int i = tid; i < ROWS * TOT; i += 256) zsel[i] = 0.0f;
  if (tid < ROWS) lens[tid] = p.len[rb + tid];

  v8f c_t = bcast8(0.0f);   // text cell state, per-wave columns [16w,16w+16)
  v8f c_s = bcast8(0.0f);   // audio (waves 0,1) / video (waves 2,3) cell state
  v8f h_t = bcast8(0.0f), h_s = bcast8(0.0f);

  __syncthreads();

  for (int t = 0; t < TSTEPS; ++t) {
    // -------- phase 1: stage x tiles (contiguous 16-row blocks) --------
    {
      const float* gt = p.xt + ((size_t)t * NBATCH + rb) * DT;
      for (int i = tid; i < ROWS * DT; i += 256) xt[(i / DT) * SXT + (i % DT)] = gt[i];
      const float* ga = p.xa + ((size_t)t * NBATCH + rb) * DA;
      for (int i = tid; i < ROWS * DA; i += 256) xa[(i / DA) * SXA + (i % DA)] = ga[i];
      const float* gv = p.xv + ((size_t)t * NBATCH + rb) * DV;
      for (int i = tid; i < ROWS * DV; i += 256) xv[(i / DV) * SXV + (i % DV)] = gv[i];
      if (t + 1 < TSTEPS) {   // hide next step's 19KB text fetch behind WMMA work
        const float* nt = p.xt + ((size_t)(t + 1) * NBATCH + rb) * DT;
        for (int i = tid * 16; i < ROWS * DT; i += 256 * 16) __builtin_prefetch(nt + i, 0, 0);
      }
    }
    __syncthreads();

    // -------- phase 2: LSTHM cells (reads hc/zb/x; state in registers) --------
    lsthm_cell(xt, SXT, DT / 4, p.Wt, hc, HT / 4, p.Ut, zb, p.Vt,
               p.Wb_t, p.Ub_t, p.Vb_t, HT, 16 * w, lane, lrow, half, c_t, h_t);
    if (w < 2) {
      lsthm_cell(xa, SXA, 76 / 4, p.Wa, hc + HT, HA / 4, p.Ua, zb, p.Va,
                 p.Wb_a, p.Ub_a, p.Vb_a, HA, 16 * w, lane, lrow, half, c_s, h_s);
    } else if (w < 4) {
      lsthm_cell(xv, SXV, 36 / 4, p.Wv, hc + HT + HA, HV / 4, p.Uv, zb, p.Vv,
                 p.Wb_v, p.Ub_v, p.Vb_v, HV, 16 * (w - 2), lane, lrow, half, c_s, h_s);
    }
    __syncthreads();   // all reads of old hc complete

    // -------- phase 3: publish h into hc --------
#pragma unroll
    for (int v = 0; v < 8; ++v)
      hc[(v + 8 * half) * SHC + 16 * w + lrow] = h_t[v];
    if (w < 2) {
#pragma unroll
      for (int v = 0; v < 8; ++v)
        hc[(v + 8 * half) * SHC + HT + 16 * w + lrow] = h_s[v];
    } else if (w < 4) {
#pragma unroll
      for (int v = 0; v < 8; ++v)
        hc[(v + 8 * half) * SHC + HT + HA + 16 * (w - 2) + lrow] = h_s[v];
    }
    __syncthreads();

    // -------- phase 4: attention scores  att = hc @ att_W + att_b  [16x768] --------
    for (int j = 0; j < 6; j += 2) {
      const int ca = 16 * w + 128 * j;
      const int cb = 16 * w + 128 * (j + 1);
      v8f A0 = bcast8(p.att_b[ca + lrow]);
      v8f A1 = bcast8(p.att_b[cb + lrow]);
      gemm2p(hc, SHC, IdMap(), p.AW, TOT / 4, ca >> 4, cb >> 4, lane, lrow, half, A0, A1);
#pragma unroll
      for (int v = 0; v < 8; ++v) {
        att[(v + 8 * half) * SAT + ca + lrow] = A0[v];
        att[(v + 8 * half) * SAT + cb + lrow] = A1[v];
      }
    }
    __syncthreads();

    // -------- phase 5: softmax over TOT per (row,k), times hc -> kh in place --------
    if (tid < ROWS * KATT) {
      const int r = tid >> 2, kk = tid & 3;
      float* row = att + r * SAT + kk * TOT;
      const float* hrow = hc + r * SHC;
      float m = -3.0e38f;
      for (int c = 0; c < TOT; ++c) m = fmaxf(m, row[c]);
      float s = 0.0f;
      for (int c = 0; c < TOT; ++c) { float e = __expf(row[c] - m); row[c] = e; s += e; }
      const float inv = 1.0f / s;
      for (int c = 0; c < TOT; ++c) row[c] = row[c] * inv * hrow[c];
    }
    __syncthreads();

    // -------- phase 6: strided-view projections  s_t/s_a/s_v -> sloc[16x96] --------
    if (w < 4) {                        // s_t: [16,512] @ R_t[512,64]
      const int c0 = 16 * w;
      v8f A0 = bcast8(p.Rb_t[c0 + lrow]);
      gemm1p(att, SAT, MapT(), p.Rt, (KATT * HT) / 4, w, lane, lrow, half, A0);
#pragma unroll
      for (int v = 0; v < 8; ++v) sloc[(v + 8 * half) * SSL + c0 + lrow] = A0[v];
    } else if (w == 4) {                // s_a: [16,128] @ R_a[128,16]
      v8f A0 = bcast8(p.Rb_a[lrow]);
      gemm1p(att, SAT, MapA(), p.Ra, (KATT * HA) / 4, 0, lane, lrow, half, A0);
#pragma unroll
      for (int v = 0; v < 8; ++v) sloc[(v + 8 * half) * SSL + 64 + lrow] = A0[v];
    } else if (w == 5) {                // s_v: [16,128] @ R_v[128,16]
      v8f A0 = bcast8(p.Rb_v[lrow]);
      gemm1p(att, SAT, MapV(), p.Rv, (KATT * HV) / 4, 0, lane, lrow, half, A0);
#pragma unroll
      for (int v = 0; v < 8; ++v) sloc[(v + 8 * half) * SSL + 80 + lrow] = A0[v];
    }
    __syncthreads();

    // -------- phase 7: z_new = sloc @ hyb_W + hyb_b  [16x192] --------
    {
      const int c0 = 16 * w;
      v8f A0 = bcast8(p.hyb_b[c0 + lrow]);
      gemm1p(sloc, SSL, IdMap(), p.HY, 96 / 4, w, lane, lrow, half, A0);
#pragma unroll
      for (int v = 0; v < 8; ++v) zb[(v + 8 * half) * SZ + c0 + lrow] = A0[v];
      if (w < 4) {
        const int cc = 128 + 16 * w;
        v8f A1 = bcast8(p.hyb_b[cc + lrow]);
        gemm1p(sloc, SSL, IdMap(), p.HY, 96 / 4, 8 + w, lane, lrow, half, A1);
#pragma unroll
        for (int v = 0; v < 8; ++v) zb[(v + 8 * half) * SZ + cc + lrow] = A1[v];
      }
    }
    __syncthreads();

    // -------- phase 8: capture z at t == lengths-1 --------
    for (int i = tid; i < ROWS * TOT; i += 256) {
      const int r = i / TOT, c = i % TOT;
      if (lens[r] - 1 == t) zsel[i] = zb[r * SZ + c];
    }
    __syncthreads();
  }

  // -------- epilogue: out = z_T @ out_W + out_b --------
  if (tid < ROWS) {
    float s = p.out_b[0];
    const float* zr = zsel + tid * TOT;
    for (int c = 0; c < TOT; ++c) s += zr[c] * p.out_W[c];
    p.out[rb + tid] = s;
  }
}

static inline void pack_launch(const float* src, float* wsf, unsigned off,
                               int K, int N, int PK, hipStream_t stream) {
  const int total = N * PK;
  pack_b_kernel<<<(total + 255) / 256, 256, 0, stream>>>(src, wsf + off, K, N, PK);
}

extern "C" void kernel_launch(void* const* d_in, const int* in_sizes, int n_in,
                              void* d_out, int out_size, void* d_ws, size_t ws_size,
                              hipStream_t stream) {
  (void)in_sizes; (void)n_in; (void)out_size; (void)ws_size;
  const float* in[34];
  for (int i = 0; i < 34; ++i) in[i] = (const float*)d_in[i];
  float* wsf = (float*)d_ws;

  // ---- pre-pack every weight matrix into WMMA B-fragment order ----
  pack_launch(in[4],  wsf, OFF_WT, DT,        4 * HT, DT,  stream);  // W_t
  pack_launch(in[6],  wsf, OFF_UT, HT,        4 * HT, HT,  stream);  // U_t
  pack_launch(in[8],  wsf, OFF_VT, TOT,       4 * HT, TOT, stream);  // V_t
  pack_launch(in[12], wsf, OFF_WA, DA,        4 * HA, 76,  stream);  // W_a (pad K->76)
  pack_launch(in[14], wsf, OFF_UA, HA,        4 * HA, HA,  stream);  // U_a
  pack_launch(in[16], wsf, OFF_VA, TOT,       4 * HA, TOT, stream);  // V_a
  pack_launch(in[20], wsf, OFF_WV, DV,        4 * HV, 36,  stream);  // W_v (pad K->36)
  pack_launch(in[22], wsf, OFF_UV, HV,        4 * HV, HV,  stream);  // U_v
  pack_launch(in[24], wsf, OFF_VV, TOT,       4 * HV, TOT, stream);  // V_v
  pack_launch(in[28], wsf, OFF_AW, TOT,       KATT * TOT, TOT, stream);  // att_W
  pack_launch(in[10], wsf, OFF_RT, KATT * HT, 64,  KATT * HT, stream);   // R_t
  pack_launch(in[18], wsf, OFF_RA, KATT * HA, 16,  KATT * HA, stream);   // R_a
  pack_launch(in[26], wsf, OFF_RV, KATT * HV, 16,  KATT * HV, stream);   // R_v
  pack_launch(in[30], wsf, OFF_HY, 96,        TOT, 96, stream);          // hyb_W

  P p;
  p.xt  = in[0]; p.xa = in[1]; p.xv = in[2];
  p.len = (const int*)d_in[3];
  p.Wt = (const v2f*)(wsf + OFF_WT); p.Ut = (const v2f*)(wsf + OFF_UT);
  p.Vt = (const v2f*)(wsf + OFF_VT);
  p.Wa = (const v2f*)(wsf + OFF_WA); p.Ua = (const v2f*)(wsf + OFF_UA);
  p.Va = (const v2f*)(wsf + OFF_VA);
  p.Wv = (const v2f*)(wsf + OFF_WV); p.Uv = (const v2f*)(wsf + OFF_UV);
  p.Vv = (const v2f*)(wsf + OFF_VV);
  p.AW = (const v2f*)(wsf + OFF_AW);
  p.Rt = (const v2f*)(wsf + OFF_RT); p.Ra = (const v2f*)(wsf + OFF_RA);
  p.Rv = (const v2f*)(wsf + OFF_RV);
  p.HY = (const v2f*)(wsf + OFF_HY);
  p.Wb_t = in[5];  p.Ub_t = in[7];  p.Vb_t = in[9];  p.Rb_t = in[11];
  p.Wb_a = in[13]; p.Ub_a = in[15]; p.Vb_a = in[17]; p.Rb_a = in[19];
  p.Wb_v = in[21]; p.Ub_v = in[23]; p.Vb_v = in[25]; p.Rb_v = in[27];
  p.att_b = in[29]; p.hyb_b = in[31]; p.out_W = in[32]; p.out_b = in[33];
  p.out = (float*)d_out;

  marn_fused_kernel<<<NBATCH / ROWS, 256, 0, stream>>>(p);
}